// GCNClassifier_74509092651321
// MI455X (gfx1250) — compile-verified
//
#include <hip/hip_runtime.h>
#include <hip/hip_bf16.h>

typedef __attribute__((ext_vector_type(2))) float v2f;
typedef __attribute__((ext_vector_type(8))) float v8f;

#define HID 64
#define NCLASSES 10
#define NGRAPHS 64

// ---------------- utility ----------------
__global__ void k_zero_f32(float* __restrict__ p, int n) {
  int i = blockIdx.x * blockDim.x + threadIdx.x;
  if (i < n) p[i] = 0.0f;
}
__global__ void k_zero_i32(int* __restrict__ p, int n) {
  int i = blockIdx.x * blockDim.x + threadIdx.x;
  if (i < n) p[i] = 0;
}

// ---------------- degree histograms (int atomics, 3.2M total) ----------------
__global__ void k_hist(const int* __restrict__ src, const int* __restrict__ dst,
                       int* __restrict__ cnt_out, int* __restrict__ cnt_in, int E) {
  int e = blockIdx.x * blockDim.x + threadIdx.x;
  if (e >= E) return;
  atomicAdd(&cnt_out[src[e]], 1);
  atomicAdd(&cnt_in[dst[e]], 1);
}

__global__ void k_norms(const int* __restrict__ cnt_out, const int* __restrict__ cnt_in,
                        float* __restrict__ ns, float* __restrict__ nd, int N) {
  int i = blockIdx.x * blockDim.x + threadIdx.x;
  if (i >= N) return;
  ns[i] = 1.0f / sqrtf(fmaxf((float)cnt_out[i], 1.0f));
  nd[i] = 1.0f / sqrtf(fmaxf((float)cnt_in[i], 1.0f));
}

// ---------------- exclusive scan of cnt_in -> offs (two-level, LDS) ----------------
__global__ void k_scan1(const int* __restrict__ cnt, int* __restrict__ offs,
                        int* __restrict__ blksum, int N) {
  __shared__ int sh[256];
  int tid = threadIdx.x;
  int gid = blockIdx.x * 256 + tid;
  int v = (gid < N) ? cnt[gid] : 0;
  sh[tid] = v;
  __syncthreads();
  for (int d = 1; d < 256; d <<= 1) {
    int t = (tid >= d) ? sh[tid - d] : 0;
    __syncthreads();
    if (tid >= d) sh[tid] += t;
    __syncthreads();
  }
  if (gid < N) offs[gid] = sh[tid] - v;   // exclusive within block
  if (tid == 255) blksum[blockIdx.x] = sh[255];
}

// single block; requires nb <= 512 (N <= 131072; here N = 100000 -> nb = 391)
__global__ void k_scan2(int* __restrict__ blksum, int nb) {
  __shared__ int sh[512];
  int tid = threadIdx.x;
  int v = (tid < nb) ? blksum[tid] : 0;
  sh[tid] = v;
  __syncthreads();
  for (int d = 1; d < 512; d <<= 1) {
    int t = (tid >= d) ? sh[tid - d] : 0;
    __syncthreads();
    if (tid >= d) sh[tid] += t;
    __syncthreads();
  }
  if (tid < nb) blksum[tid] = sh[tid] - v; // exclusive block offsets
}

__global__ void k_scan3(int* __restrict__ offs, const int* __restrict__ blksum, int N, int E) {
  int gid = blockIdx.x * blockDim.x + threadIdx.x;
  if (gid < N) offs[gid] += blksum[gid >> 8];
  if (gid == 0) offs[N] = E;
}

// ---------------- CSR placement (1.6M placement atomics only) ----------------
__global__ void k_place(const int* __restrict__ src, const int* __restrict__ dst,
                        const int* __restrict__ offs, int* __restrict__ cursor,
                        int* __restrict__ csr, int E) {
  int e = blockIdx.x * blockDim.x + threadIdx.x;
  if (e >= E) return;
  int d = dst[e];
  int p = atomicAdd(&cursor[d], 1);
  csr[offs[d] + p] = src[e];
}

// ---------------- layer 1 (rank-1 features), gather form ----------------
// agg1[n] = sum_{s in in(n)} in_deg[s] * norm_src[s]
__global__ void k_gather1(const int* __restrict__ offs, const int* __restrict__ csr,
                          const int* __restrict__ cnt_in, const float* __restrict__ ns,
                          float* __restrict__ agg1, int N) {
  int n = blockIdx.x * blockDim.x + threadIdx.x;
  if (n >= N) return;
  int b = offs[n], e = offs[n + 1];
  float a = 0.0f;
  for (int i = b; i < e; ++i) {
    int s = csr[i];
    a += (float)cnt_in[s] * ns[s];
  }
  agg1[n] = a;
}

// h[n][j] = relu(agg1[n]*nd[n]*W1[j] + b1[j])   (W1 is 1x64)
__global__ void k_layer1(const float* __restrict__ agg1, const float* __restrict__ nd,
                         const float* __restrict__ W1, const float* __restrict__ b1,
                         float* __restrict__ h, int N) {
  int idx = blockIdx.x * blockDim.x + threadIdx.x;
  if (idx >= N * HID) return;
  int n = idx >> 6, j = idx & 63;
  float v = agg1[n] * nd[n] * W1[j] + b1[j];
  h[idx] = fmaxf(v, 0.0f);
}

// ---------------- 64-wide aggregation as a gather (no atomics) ----------------
// wave per destination node; lane covers channels 2*lane, 2*lane+1
__global__ void k_gather64(const int* __restrict__ offs, const int* __restrict__ csr,
                           const float* __restrict__ h, const float* __restrict__ ns,
                           float* __restrict__ agg, int N) {
  int wave = threadIdx.x >> 5;
  int lane = threadIdx.x & 31;
  int n = blockIdx.x * 8 + wave;
  if (n >= N) return;
  int b = offs[n], e = offs[n + 1];
  const int c = lane * 2;
  float ax = 0.0f, ay = 0.0f;
  for (int i = b; i < e; ++i) {
    int s = csr[i];                 // same address across wave -> broadcast
    float w = ns[s];
    const float2 v = *reinterpret_cast<const float2*>(h + (size_t)s * HID + c);
    ax += v.x * w;
    ay += v.y * w;
  }
  float2 out; out.x = ax; out.y = ay;
  *reinterpret_cast<float2*>(agg + (size_t)n * HID + c) = out;
}

// ---------------- WMMA GEMM: h = relu((agg * nd[row]) @ W + b) ----------------
// block = 128 threads (4 waves); 16 rows/block; wave w -> output cols [16w,16w+16).
// W (16KB) staged in LDS once per block; V_WMMA_F32_16X16X4_F32, K=64 loop.
__global__ void k_gcn_gemm(const float* __restrict__ agg, const float* __restrict__ nd,
                           const float* __restrict__ W, const float* __restrict__ b,
                           float* __restrict__ hout, int N) {
  __shared__ float Wl[HID * HID];   // 16 KB of the 320 KB/WGP LDS
  const int rowbase = blockIdx.x * 16;
  if (rowbase >= N) return;         // block-uniform

  {
    const float4* Wg4 = reinterpret_cast<const float4*>(W);
    float4* Wl4 = reinterpret_cast<float4*>(Wl);
    for (int i = threadIdx.x; i < HID * HID / 4; i += blockDim.x)
      Wl4[i] = Wg4[i];              // coalesced b128 global -> ds_store_b128
  }
  __syncthreads();

  const int wave = threadIdx.x >> 5;
  const int lane = threadIdx.x & 31;
  const int half = lane >> 4;       // 0: lanes 0-15, 1: lanes 16-31
  const int m = lane & 15;          // M index for A; N index for B/C/D
  const int colg = wave * 16 + m;   // global output column

  int row = rowbase + m;
  int rowc = row < N ? row : N - 1; // clamp loads: keep EXEC all-1s for WMMA
  const float scale = nd[rowc];
  const float* arow = agg + (size_t)rowc * HID;

  v8f acc = {};
#pragma unroll
  for (int k0 = 0; k0 < HID; k0 += 4) {
    const int ka = k0 + half * 2;
    v2f a;                          // A frag: lane m holds A[m][ka], A[m][ka+1]
    a.x = arow[ka] * scale;
    a.y = arow[ka + 1] * scale;
    v2f bb;                         // B frag from LDS: W[ka][colg], W[ka+1][colg]
    bb.x = Wl[ka * HID + colg];
    bb.y = Wl[(ka + 1) * HID + colg];
    acc = __builtin_amdgcn_wmma_f32_16x16x4_f32(false, a, false, bb, (short)0, acc,
                                                false, false);
  }

  const float bias = b[colg];
#pragma unroll
  for (int i = 0; i < 8; ++i) {
    int orow = rowbase + (half ? (i + 8) : i);   // C/D: VGPR i -> M=i / M=i+8
    if (orow < N)
      hout[(size_t)orow * HID + colg] = fmaxf(acc[i] + bias, 0.0f);
  }
}

// ---------------- graph pooling (graph_ids sorted -> run-flush atomics) ----------------
__global__ void k_counts(const int* __restrict__ gids, float* __restrict__ counts, int N) {
  int i = blockIdx.x * blockDim.x + threadIdx.x;
  if (i >= N) return;
  atomicAdd(&counts[gids[i]], 1.0f);
}

// block covers 64 consecutive nodes; thread (r,c) sums rows r, r+4, ... for channel c,
// flushing one atomic per graph transition (sorted gids -> ~1-2 flushes/thread)
__global__ void k_pool(const int* __restrict__ gids, const float* __restrict__ h,
                       float* __restrict__ hg, int N) {
  int n0 = blockIdx.x * 64;
  int c = threadIdx.x & 63;
  int r = threadIdx.x >> 6;           // 0..3
  float acc = 0.0f;
  int curg = -1;
  for (int i = r; i < 64; i += 4) {
    int n = n0 + i;
    if (n >= N) break;
    int g = gids[n];
    if (g != curg) {
      if (curg >= 0) atomicAdd(&hg[curg * HID + c], acc);
      acc = 0.0f;
      curg = g;
    }
    acc += h[(size_t)n * HID + c];
  }
  if (curg >= 0) atomicAdd(&hg[curg * HID + c], acc);
}

// single block: hg /= counts, then logits = hg @ Wc + bc
__global__ void k_final(float* __restrict__ hg, const float* __restrict__ counts,
                        const float* __restrict__ Wc, const float* __restrict__ bc,
                        float* __restrict__ logits) {
  for (int i = threadIdx.x; i < NGRAPHS * HID; i += blockDim.x)
    hg[i] = hg[i] / fmaxf(counts[i >> 6], 1.0f);
  __syncthreads();
  for (int i = threadIdx.x; i < NGRAPHS * NCLASSES; i += blockDim.x) {
    int g = i / NCLASSES, c = i % NCLASSES;
    float s = bc[c];
    for (int j = 0; j < HID; ++j) s += hg[g * HID + j] * Wc[j * NCLASSES + c];
    logits[i] = s;
  }
}

// ---------------- host launcher ----------------
extern "C" void kernel_launch(void* const* d_in, const int* in_sizes, int n_in,
                              void* d_out, int out_size, void* d_ws, size_t ws_size,
                              hipStream_t stream) {
  const int* src  = (const int*)d_in[0];
  const int* dst  = (const int*)d_in[1];
  const int* gids = (const int*)d_in[2];
  // d_in[3] = n_graphs scalar (known 64)
  const float* W1 = (const float*)d_in[4];
  const float* b1 = (const float*)d_in[5];
  const float* W2 = (const float*)d_in[6];
  const float* b2 = (const float*)d_in[7];
  const float* W3 = (const float*)d_in[8];
  const float* b3 = (const float*)d_in[9];
  const float* W4 = (const float*)d_in[10];
  const float* b4 = (const float*)d_in[11];
  const float* Wc = (const float*)d_in[12];
  const float* bc = (const float*)d_in[13];
  const int E = in_sizes[0];
  const int N = in_sizes[2];

  // ---- workspace layout: int region then float region (16B-aligned seam) ----
  int* cnt_in  = (int*)d_ws;                  // N
  int* cnt_out = cnt_in + N;                  // N
  int* cursor  = cnt_out + N;                 // N   (zeroed together with cnt_*)
  int* offs    = cursor + N;                  // N+1
  int* blksum  = offs + (N + 1);              // 512
  int* csr     = blksum + 512;                // E
  size_t nint  = 4 * (size_t)N + 1 + 512 + (size_t)E;
  nint = (nint + 3) & ~(size_t)3;             // pad to 16B boundary
  float* ns    = (float*)d_ws + nint;         // N
  float* nd    = ns + N;                      // N
  float* agg1  = nd + N;                      // N
  float* h     = agg1 + N;                    // N*64 (16B aligned: nint%4==0, 3N%4==0)
  float* agg   = h + (size_t)N * HID;         // N*64
  float* counts = agg + (size_t)N * HID;      // 64

  float* hg     = (float*)d_out;              // 64*64
  float* logits = hg + NGRAPHS * HID;         // 64*10

  const int B = 256;
  auto cdiv = [](long long a, long long b) { return (int)((a + b - 1) / b); };
  const int nb = cdiv(N, 256);                // scan blocks (391 <= 512)

  // degrees + norms + CSR build
  k_zero_i32<<<cdiv(3LL * N, B), B, 0, stream>>>(cnt_in, 3 * N); // cnt_in,cnt_out,cursor
  k_hist<<<cdiv(E, B), B, 0, stream>>>(src, dst, cnt_out, cnt_in, E);
  k_norms<<<cdiv(N, B), B, 0, stream>>>(cnt_out, cnt_in, ns, nd, N);
  k_scan1<<<nb, 256, 0, stream>>>(cnt_in, offs, blksum, N);
  k_scan2<<<1, 512, 0, stream>>>(blksum, nb);
  k_scan3<<<cdiv(N, B), B, 0, stream>>>(offs, blksum, N, E);
  k_place<<<cdiv(E, B), B, 0, stream>>>(src, dst, offs, cursor, csr, E);

  // layer 1
  k_gather1<<<cdiv(N, B), B, 0, stream>>>(offs, csr, cnt_in, ns, agg1, N);
  k_layer1<<<cdiv((long long)N * HID, B), B, 0, stream>>>(agg1, nd, W1, b1, h, N);

  // layers 2..4: atomic-free gather + WMMA GEMM
  const float* Wlw[3] = {W2, W3, W4};
  const float* blw[3] = {b2, b3, b4};
  for (int l = 0; l < 3; ++l) {
    k_gather64<<<cdiv(N, 8), 256, 0, stream>>>(offs, csr, h, ns, agg, N);
    k_gcn_gemm<<<cdiv(N, 16), 128, 0, stream>>>(agg, nd, Wlw[l], blw[l], h, N);
  }

  // pooling + classifier
  k_zero_f32<<<1, B, 0, stream>>>(counts, NGRAPHS);
  k_zero_f32<<<cdiv(NGRAPHS * (HID + NCLASSES), B), B, 0, stream>>>(
      hg, NGRAPHS * (HID + NCLASSES));
  k_counts<<<cdiv(N, B), B, 0, stream>>>(gids, counts, N);
  k_pool<<<cdiv(N, 64), 256, 0, stream>>>(gids, h, hg, N);
  k_final<<<1, 256, 0, stream>>>(hg, counts, Wc, bc, logits);
}